// MMTCBlock_2757369004154
// MI455X (gfx1250) — compile-verified
//
#include <hip/hip_runtime.h>
#include <math.h>

typedef float v2f __attribute__((ext_vector_type(2)));
typedef float v8f __attribute__((ext_vector_type(8)));

#define DIMC 512   // channels
#define NTOK 512   // tokens per modality (8*8*8)
#define NB   16    // batch
#define NMOD 4     // modalities

// CDNA5 async global->LDS copies (ASYNCcnt-tracked, no VGPR staging).
#define USE_ASYNC_LDS 1

enum { A_ROW = 0, A_TRANS = 1, A_CONCAT = 2 };
enum { E_STORE = 0, E_BIAS = 1, E_BIAS_GELU = 2, E_BIAS_TSTORE = 3 };

struct MlpPtrs {
  const float* w1[16];
  const float* b1[16];
  const float* w2[16];
  const float* b2[16];
};

__device__ __forceinline__ int triple_slot(const int* mask, int b, int i, int j) {
  // enumeration order: missing i ascending, avail j ascending (max 4 triples per b)
  int t = 0;
  for (int ii = 0; ii < 4; ++ii) {
    if (mask[b * 4 + ii]) continue;
    for (int jj = 0; jj < 4; ++jj) {
      if (!mask[b * 4 + jj]) continue;
      if (ii == i && jj == j) return t;
      ++t;
    }
  }
  return 0;
}

// one element global -> LDS; async path uses GLOBAL_LOAD_ASYNC_TO_LDS_B32
// (flat shared pointer low 32 bits == LDS byte address per aperture rules)
__device__ __forceinline__ void g2lds_b32(const float* __restrict__ gsrc,
                                          float* __restrict__ ldst) {
#if USE_ASYNC_LDS
  asm volatile("global_load_async_to_lds_b32 %0, %1, off"
               :
               : "v"((unsigned)(size_t)ldst), "v"(gsrc)
               : "memory");
#else
  *ldst = *gsrc;
#endif
}

// ASYNCcnt retires in order: each stage issues exactly 24 async ops per lane,
// so waiting "<= 24" after issuing the next-buffer prefetch guarantees the
// CURRENT buffer is complete while the prefetch stays in flight.
__device__ __forceinline__ void g2lds_wait(bool prefetch_in_flight) {
#if USE_ASYNC_LDS
  if (prefetch_in_flight) asm volatile("s_wait_asynccnt 24" ::: "memory");
  else                    asm volatile("s_wait_asynccnt 0" ::: "memory");
#else
  (void)prefetch_in_flight;
#endif
}

// ---------------------------------------------------------------------------
// Core 128x64-tile fp32 WMMA GEMM: D = op(A)[M,K] @ B[K,N] (+bias/gelu/tstore)
// 256 threads = 8 waves (wave32); waves in 4x2 grid, each owns a 32x32 slice
// (4 accumulators) so 8 LDS dwords/lane feed 4 V_WMMA_F32_16X16X4_F32 ops.
// Double-buffered LDS staging overlaps async global->LDS with WMMA compute.
// ---------------------------------------------------------------------------
template <int AL, int EP>
__device__ __forceinline__ void gemm128(
    const float* __restrict__ a0, const float* __restrict__ a1,
    const float* __restrict__ a2, const float* __restrict__ a3, int lda,
    const float* __restrict__ B, int ldb,
    float* __restrict__ D, int ldd,
    const float* __restrict__ bias,
    int mTile, int nTile, int K) {
  __shared__ union {
    struct { float As[128][33]; float Bs[32][65]; } ld[2];
    float Ds[128][65];
  } sm;

  const int tid  = threadIdx.x;
  const int wave = tid >> 5;
  const int lane = tid & 31;
  const int wm   = (wave & 3) * 32;   // wave row offset within tile
  const int wn   = (wave >> 2) * 32;  // wave col offset within tile
  const int half = lane >> 4;         // K-pair select for f32 WMMA A/B frags
  const int l15  = lane & 15;

  v8f acc00 = {0.f, 0.f, 0.f, 0.f, 0.f, 0.f, 0.f, 0.f};
  v8f acc01 = acc00, acc10 = acc00, acc11 = acc00;

  // stage one 128x32 A tile + 32x64 B tile into LDS buffer `buf`
  auto stage = [&](int buf, int k0) {
#pragma unroll
    for (int it = 0; it < 16; ++it) {
      int idx = it * 256 + tid;
      int m, k;
      if (AL == A_TRANS) { m = idx & 127; k = idx >> 7; }  // coalesce along m
      else               { m = idx >> 5;  k = idx & 31; }  // coalesce along k
      int gk = k0 + k;
      const float* src;
      if (AL == A_ROW) {
        src = &a0[(size_t)(mTile + m) * lda + gk];
      } else if (AL == A_TRANS) {
        src = &a0[(size_t)gk * lda + (mTile + m)];
      } else {  // A_CONCAT: 512-wide column segments from separate buffers
        int seg = gk >> 9;
        const float* base = (seg == 0) ? a0 : (seg == 1) ? a1 : (seg == 2) ? a2 : a3;
        src = &base[(size_t)(mTile + m) * DIMC + (gk & 511)];
      }
      g2lds_b32(src, &sm.ld[buf].As[m][k]);
    }
#pragma unroll
    for (int it = 0; it < 8; ++it) {
      int idx = it * 256 + tid;
      int k = idx >> 6;
      int n = idx & 63;
      g2lds_b32(&B[(size_t)(k0 + k) * ldb + (nTile + n)], &sm.ld[buf].Bs[k][n]);
    }
  };

  const int nchunk = K >> 5;
  stage(0, 0);
  for (int c = 0; c < nchunk; ++c) {
    const int cur = c & 1;
    const bool hasNext = (c + 1) < nchunk;
    if (hasNext) stage(cur ^ 1, (c + 1) << 5);  // prefetch next chunk
    g2lds_wait(hasNext);                        // current buffer complete
    __syncthreads();
    // --- 8 x (K=4) WMMA steps, 4 WMMAs each ---
#pragma unroll
    for (int kk = 0; kk < 32; kk += 4) {
      v2f af0, af1, bf0, bf1;
      af0.x = sm.ld[cur].As[wm + l15][kk + 2 * half];
      af0.y = sm.ld[cur].As[wm + l15][kk + 2 * half + 1];
      af1.x = sm.ld[cur].As[wm + 16 + l15][kk + 2 * half];
      af1.y = sm.ld[cur].As[wm + 16 + l15][kk + 2 * half + 1];
      bf0.x = sm.ld[cur].Bs[kk + 2 * half][wn + l15];
      bf0.y = sm.ld[cur].Bs[kk + 2 * half + 1][wn + l15];
      bf1.x = sm.ld[cur].Bs[kk + 2 * half][wn + 16 + l15];
      bf1.y = sm.ld[cur].Bs[kk + 2 * half + 1][wn + 16 + l15];
      acc00 = __builtin_amdgcn_wmma_f32_16x16x4_f32(false, af0, false, bf0, (short)0, acc00, false, false);
      acc01 = __builtin_amdgcn_wmma_f32_16x16x4_f32(false, af0, false, bf1, (short)0, acc01, false, false);
      acc10 = __builtin_amdgcn_wmma_f32_16x16x4_f32(false, af1, false, bf0, (short)0, acc10, false, false);
      acc11 = __builtin_amdgcn_wmma_f32_16x16x4_f32(false, af1, false, bf1, (short)0, acc11, false, false);
    }
    __syncthreads();  // all waves done reading `cur` before it is restaged
  }

  if (EP == E_BIAS_TSTORE) {
    // deposit to LDS, then coalesced transposed store: D[(n)*ldd + m]
#pragma unroll
    for (int r = 0; r < 8; ++r) {
      int mr = half * 8 + r;
      int n0 = wn + l15, n1 = wn + 16 + l15;
      sm.Ds[wm + mr][n0]      = acc00[r] + bias[nTile + n0];
      sm.Ds[wm + mr][n1]      = acc01[r] + bias[nTile + n1];
      sm.Ds[wm + 16 + mr][n0] = acc10[r] + bias[nTile + n0];
      sm.Ds[wm + 16 + mr][n1] = acc11[r] + bias[nTile + n1];
    }
    __syncthreads();
#pragma unroll
    for (int it = 0; it < 32; ++it) {
      int idx = it * 256 + tid;
      int c = idx >> 7;    // 0..63
      int m = idx & 127;   // 0..127
      D[(size_t)(nTile + c) * ldd + (mTile + m)] = sm.Ds[m][c];
    }
  } else {
#pragma unroll
    for (int r = 0; r < 8; ++r) {
      int mr = half * 8 + r;
      int n0 = nTile + wn + l15;
      int n1 = nTile + wn + 16 + l15;
      float v00 = acc00[r], v01 = acc01[r], v10 = acc10[r], v11 = acc11[r];
      if (EP == E_BIAS || EP == E_BIAS_GELU) {
        v00 += bias[n0]; v01 += bias[n1]; v10 += bias[n0]; v11 += bias[n1];
      }
      if (EP == E_BIAS_GELU) {
        v00 = 0.5f * v00 * (1.f + erff(v00 * 0.70710678118654752440f));
        v01 = 0.5f * v01 * (1.f + erff(v01 * 0.70710678118654752440f));
        v10 = 0.5f * v10 * (1.f + erff(v10 * 0.70710678118654752440f));
        v11 = 0.5f * v11 * (1.f + erff(v11 * 0.70710678118654752440f));
      }
      size_t m0 = (size_t)(mTile + wm + mr) * ldd;
      size_t m1 = (size_t)(mTile + wm + 16 + mr) * ldd;
      D[m0 + n0] = v00;
      D[m0 + n1] = v01;
      D[m1 + n0] = v10;
      D[m1 + n1] = v11;
    }
  }
}

// ---------------------------------------------------------------------------
// Small helper kernels (non-GEMM parts are <1% of FLOPs)
// ---------------------------------------------------------------------------

__global__ void k_set_loss(float* loss) { *loss = 0.f; }

// Per modality i: q_i = agent_tokens[i] @ uca_q[i]; Wsc_i[c][h*8+t] = sum_d uca_k[i][c][h64+d]*q[t][h64+d]
__global__ void k_uca_prep(const float* __restrict__ at, const float* __restrict__ ucaq,
                           const float* __restrict__ ucak, float* __restrict__ wsc,
                           float* loss) {
  int i = blockIdx.x;
  int tid = threadIdx.x;
  if (i == 0 && tid == 0) *loss = 0.f;
  __shared__ float q[8][512];
  const float* A  = at + (size_t)i * 8 * DIMC;
  const float* Wq = ucaq + (size_t)i * DIMC * DIMC;
  for (int o = tid; o < 8 * DIMC; o += 256) {
    int t = o >> 9, c = o & 511;
    float s = 0.f;
    for (int k = 0; k < DIMC; ++k) s += A[t * DIMC + k] * Wq[(size_t)k * DIMC + c];
    q[t][c] = s;
  }
  __syncthreads();
  const float* Wk = ucak + (size_t)i * DIMC * DIMC;
  float* W = wsc + (size_t)i * DIMC * 64;
  for (int o = tid; o < DIMC * 64; o += 256) {
    int c = o >> 6, ha = o & 63, h = ha >> 3, a = ha & 7;
    float s = 0.f;
    for (int d = 0; d < 64; ++d) s += Wk[(size_t)c * DIMC + h * 64 + d] * q[a][h * 64 + d];
    W[o] = s;
  }
}

__device__ __forceinline__ void col_softmax_512x64(float* __restrict__ T) {
  int col = threadIdx.x;  // 64 threads
  float mx = -1e30f;
  for (int n = 0; n < NTOK; ++n) mx = fmaxf(mx, T[n * 64 + col] * 0.125f);
  float sum = 0.f;
  for (int n = 0; n < NTOK; ++n) {
    float e = expf(T[n * 64 + col] * 0.125f - mx);
    T[n * 64 + col] = e;
    sum += e;
  }
  float inv = 1.f / sum;
  for (int n = 0; n < NTOK; ++n) T[n * 64 + col] *= inv;
}

__global__ void k_uca_softmax(const int* __restrict__ mask, float* __restrict__ T1) {
  int z = blockIdx.x, b = z >> 2, i = z & 3;
  if (!mask[b * 4 + i]) return;
  col_softmax_512x64(T1 + (size_t)z * NTOK * 64);
}

// av8[t][c] = sum_k uca_v[i][k][c]*U1[k][h(c)*8+t];  ats = av8 @ uca_pw[i] + uca_pb[i]
__global__ void k_uca_finish(const int* __restrict__ mask, const float* __restrict__ at,
                             const float* __restrict__ ucav, const float* __restrict__ ucapw,
                             const float* __restrict__ ucapb, const float* __restrict__ U1,
                             float* __restrict__ ats) {
  int z = blockIdx.x, b = z >> 2, i = z & 3;
  int tid = threadIdx.x;
  float* out = ats + (size_t)z * 8 * DIMC;
  if (!mask[b * 4 + i]) {
    const float* src = at + (size_t)i * 8 * DIMC;
    for (int o = tid; o < 8 * DIMC; o += 256) out[o] = src[o];
    return;
  }
  __shared__ float av[8][512];
  const float* U  = U1 + (size_t)z * NTOK * 64;
  const float* Wv = ucav + (size_t)i * DIMC * DIMC;
  for (int o = tid; o < 8 * DIMC; o += 256) {
    int t = o >> 9, c = o & 511, h = c >> 6;
    float s = 0.f;
    for (int k = 0; k < DIMC; ++k) s += Wv[(size_t)k * DIMC + c] * U[k * 64 + h * 8 + t];
    av[t][c] = s;
  }
  __syncthreads();
  const float* Wp = ucapw + (size_t)i * DIMC * DIMC;
  for (int o = tid; o < 8 * DIMC; o += 256) {
    int t = o >> 9, c = o & 511;
    float s = ucapb[i * DIMC + c];
    for (int k = 0; k < DIMC; ++k) s += av[t][k] * Wp[(size_t)k * DIMC + c];
    out[o] = s;
  }
}

// ac[b][i][j] = _cross(ats_i, ats_j, aca_*) : all-8x8, done directly
__global__ void k_aca_cross(const int* __restrict__ mask, const float* __restrict__ ats,
                            const float* __restrict__ acaq, const float* __restrict__ acak,
                            const float* __restrict__ acav, const float* __restrict__ acapw,
                            const float* __restrict__ acapb, float* __restrict__ ac) {
  int z = blockIdx.x, b = z >> 4, i = (z >> 2) & 3, j = z & 3;
  if (mask[b * 4 + i] || !mask[b * 4 + j]) return;
  __shared__ float q[8][512], kk[8][512], vv[8][512], oo[8][512];
  const float* qi = ats + (size_t)(b * 4 + i) * 8 * DIMC;
  const float* kj = ats + (size_t)(b * 4 + j) * 8 * DIMC;
  int tid = threadIdx.x;
  for (int o = tid; o < 8 * DIMC; o += 256) {
    int t = o >> 9, c = o & 511;
    float sq = 0.f, sk = 0.f, sv = 0.f;
    for (int k = 0; k < DIMC; ++k) {
      float xi = qi[t * DIMC + k], xj = kj[t * DIMC + k];
      sq += xi * acaq[(size_t)k * DIMC + c];
      sk += xj * acak[(size_t)k * DIMC + c];
      sv += xj * acav[(size_t)k * DIMC + c];
    }
    q[t][c] = sq; kk[t][c] = sk; vv[t][c] = sv;
  }
  __syncthreads();
  if (tid < 64) {
    int h = tid >> 3, t = tid & 7;
    float s[8], mx = -1e30f;
    for (int a = 0; a < 8; ++a) {
      float d = 0.f;
      for (int e = 0; e < 64; ++e) d += q[t][h * 64 + e] * kk[a][h * 64 + e];
      s[a] = d * 0.125f;
      mx = fmaxf(mx, s[a]);
    }
    float sum = 0.f;
    for (int a = 0; a < 8; ++a) { s[a] = expf(s[a] - mx); sum += s[a]; }
    float inv = 1.f / sum;
    for (int e = 0; e < 64; ++e) {
      float d = 0.f;
      for (int a = 0; a < 8; ++a) d += s[a] * vv[a][h * 64 + e];
      oo[t][h * 64 + e] = d * inv;
    }
  }
  __syncthreads();
  float* out = ac + (size_t)z * 8 * DIMC;
  for (int o = tid; o < 8 * DIMC; o += 256) {
    int t = o >> 9, c = o & 511;
    float s = acapb[c];
    for (int k = 0; k < DIMC; ++k) s += oo[t][k] * acapw[(size_t)k * DIMC + c];
    out[o] = s;
  }
}

// W2k/W2q[c][h*8+a] = sum_d aa_{k,q}[c][h64+d] * ac[a][h64+d]
__global__ void k_agent_prep(const int* __restrict__ mask, const float* __restrict__ ac,
                             const float* __restrict__ aaq, const float* __restrict__ aak,
                             float* __restrict__ W2q, float* __restrict__ W2k) {
  int z = blockIdx.x, b = z >> 4, i = (z >> 2) & 3, j = z & 3;
  if (mask[b * 4 + i] || !mask[b * 4 + j]) return;
  int t = triple_slot(mask, b, i, j);
  const float* a = ac + (size_t)z * 8 * DIMC;
  float* wq = W2q + (size_t)(b * 4 + t) * DIMC * 64;
  float* wk = W2k + (size_t)(b * 4 + t) * DIMC * 64;
  for (int o = threadIdx.x; o < DIMC * 64; o += 256) {
    int c = o >> 6, ha = o & 63, h = ha >> 3, aa = ha & 7;
    float sq = 0.f, sk = 0.f;
    for (int d = 0; d < 64; ++d) {
      float av_ = a[aa * DIMC + h * 64 + d];
      sq += aaq[(size_t)c * DIMC + h * 64 + d] * av_;
      sk += aak[(size_t)c * DIMC + h * 64 + d] * av_;
    }
    wq[o] = sq; wk[o] = sk;
  }
}

__global__ void k_agent_softmax_k(const int* __restrict__ mask, float* __restrict__ Sk) {
  int z = blockIdx.x, b = z >> 4, i = (z >> 2) & 3, j = z & 3;
  if (mask[b * 4 + i] || !mask[b * 4 + j]) return;
  int t = triple_slot(mask, b, i, j);
  col_softmax_512x64(Sk + (size_t)(b * 4 + t) * NTOK * 64);
}

__global__ void k_agent_softmax_q(const int* __restrict__ mask, float* __restrict__ Sq) {
  int z = blockIdx.x, b = z >> 4, i = (z >> 2) & 3, j = z & 3;
  if (mask[b * 4 + i] || !mask[b * 4 + j]) return;
  int t = triple_slot(mask, b, i, j);
  float* S = Sq + (size_t)(b * 4 + t) * NTOK * 64;
  for (int g = threadIdx.x; g < NTOK * 8; g += 256) {
    int n = g >> 3, h = g & 7;
    float* p = S + n * 64 + h * 8;
    float mx = -1e30f;
    float s[8];
    for (int a = 0; a < 8; ++a) { s[a] = p[a] * 0.125f; mx = fmaxf(mx, s[a]); }
    float sum = 0.f;
    for (int a = 0; a < 8; ++a) { s[a] = expf(s[a] - mx); sum += s[a]; }
    float inv = 1.f / sum;
    for (int a = 0; a < 8; ++a) p[a] = s[a] * inv;
  }
}

// av2[a][c] = sum_k aa_v[k][c] * U2[k][h(c)*8+a]
__global__ void k_agent_av(const int* __restrict__ mask, const float* __restrict__ aav,
                           const float* __restrict__ U2, float* __restrict__ av2) {
  int z = blockIdx.x, b = z >> 4, i = (z >> 2) & 3, j = z & 3;
  if (mask[b * 4 + i] || !mask[b * 4 + j]) return;
  int t = triple_slot(mask, b, i, j);
  const float* U = U2 + (size_t)(b * 4 + t) * NTOK * 64;
  float* out = av2 + (size_t)(b * 4 + t) * 8 * DIMC;
  for (int o = threadIdx.x; o < 8 * DIMC; o += 256) {
    int a = o >> 9, c = o & 511, h = c >> 6;
    float s = 0.f;
    for (int k = 0; k < DIMC; ++k) s += aav[(size_t)k * DIMC + c] * U[k * 64 + h * 8 + a];
    out[o] = s;
  }
}

// attended_pre[n][c] = sum_a Sq[n][h(c)*8+a] * av2[a][c]
__global__ void k_attended_pre(const int* __restrict__ mask, const float* __restrict__ Sq,
                               const float* __restrict__ av2, float* __restrict__ scrA) {
  int z = blockIdx.x, b = z >> 4, i = (z >> 2) & 3, j = z & 3;
  if (mask[b * 4 + i] || !mask[b * 4 + j]) return;
  int t = triple_slot(mask, b, i, j);
  const float* S  = Sq + (size_t)(b * 4 + t) * NTOK * 64;
  const float* av = av2 + (size_t)(b * 4 + t) * 8 * DIMC;
  float* P = scrA + (size_t)b * 1048576 + (size_t)t * 262144;
  for (int o = threadIdx.x; o < NTOK * DIMC; o += 256) {
    int n = o >> 9, c = o & 511, h = c >> 6;
    float s = 0.f;
    for (int a = 0; a < 8; ++a) s += S[n * 64 + h * 8 + a] * av[a * DIMC + c];
    P[o] = s;
  }
}

// ---------------------------------------------------------------------------
// WMMA GEMM wrapper kernels (grid.z carries routing; blocks early-exit)
// ---------------------------------------------------------------------------

__global__ void k_uca_scores(const int* __restrict__ mask, const float* r0, const float* r1,
                             const float* r2, const float* r3, const float* __restrict__ wsc,
                             float* __restrict__ T1) {
  int z = blockIdx.z, b = z >> 2, i = z & 3;
  if (!mask[b * 4 + i]) return;
  const float* rts[4] = {r0, r1, r2, r3};
  const float* A = rts[i] + (size_t)b * DIMC * NTOK;
  gemm128<A_TRANS, E_STORE>(A, A, A, A, NTOK, wsc + (size_t)i * DIMC * 64, 64,
                            T1 + (size_t)z * NTOK * 64, 64, nullptr,
                            blockIdx.y * 128, blockIdx.x * 64, DIMC);
}

__global__ void k_uca_U(const int* __restrict__ mask, const float* r0, const float* r1,
                        const float* r2, const float* r3, const float* __restrict__ T1,
                        float* __restrict__ U1) {
  int z = blockIdx.z, b = z >> 2, i = z & 3;
  if (!mask[b * 4 + i]) return;
  const float* rts[4] = {r0, r1, r2, r3};
  const float* A = rts[i] + (size_t)b * DIMC * NTOK;
  gemm128<A_ROW, E_STORE>(A, A, A, A, NTOK, T1 + (size_t)z * NTOK * 64, 64,
                          U1 + (size_t)z * NTOK * 64, 64, nullptr,
                          blockIdx.y * 128, blockIdx.x * 64, NTOK);
}

__global__ void k_agent_scores(const int* __restrict__ mask, const float* r0, const float* r1,
                               const float* r2, const float* r3, const float* __restrict__ W2k,
                               const float* __restrict__ W2q, float* __restrict__ Sk,
                               float* __restrict__ Sq) {
  int z = blockIdx.z;
  int which = z & 1;
  int combo = z >> 1;
  int b = combo >> 4, i = (combo >> 2) & 3, j = combo & 3;
  if (mask[b * 4 + i] || !mask[b * 4 + j]) return;
  int t = triple_slot(mask, b, i, j);
  const float* rts[4] = {r0, r1, r2, r3};
  const float* A = rts[j] + (size_t)b * DIMC * NTOK;
  const float* B = (which ? W2q : W2k) + (size_t)(b * 4 + t) * DIMC * 64;
  float* D = (which ? Sq : Sk) + (size_t)(b * 4 + t) * NTOK * 64;
  gemm128<A_TRANS, E_STORE>(A, A, A, A, NTOK, B, 64, D, 64, nullptr,
                            blockIdx.y * 128, blockIdx.x * 64, DIMC);
}

__global__ void k_agent_U(const int* __restrict__ mask, const float* r0, const float* r1,
                          const float* r2, const float* r3, const float* __restrict__ Sk,
                          float* __restrict__ U2) {
  int z = blockIdx.z, b = z >> 4, i = (z >> 2) & 3, j = z & 3;
  if (mask[b * 4 + i] || !mask[b * 4 + j]) return;
  int t = triple_slot(mask, b, i, j);
  const float* rts[4] = {r0, r1, r2, r3};
  const float* A = rts[j] + (size_t)b * DIMC * NTOK;
  gemm128<A_ROW, E_STORE>(A, A, A, A, NTOK, Sk + (size_t)(b * 4 + t) * NTOK * 64, 64,
                          U2 + (size_t)(b * 4 + t) * NTOK * 64, 64, nullptr,
                          blockIdx.y * 128, blockIdx.x * 64, NTOK);
}

__global__ void k_attended_gemm(const int* __restrict__ mask, const float* __restrict__ scrA,
                                const float* __restrict__ aapw, const float* __restrict__ aapb,
                                float* __restrict__ attd) {
  int z = blockIdx.z, b = z >> 4, i = (z >> 2) & 3, j = z & 3;
  if (mask[b * 4 + i] || !mask[b * 4 + j]) return;
  int t = triple_slot(mask, b, i, j);
  const float* A = scrA + (size_t)b * 1048576 + (size_t)t * 262144;
  gemm128<A_ROW, E_BIAS>(A, A, A, A, DIMC, aapw, DIMC,
                         attd + (size_t)(b * 4 + t) * 262144, DIMC, aapb,
                         blockIdx.y * 128, blockIdx.x * 64, DIMC);
}

__global__ void k_mlp1(const int* __restrict__ mask, const float* __restrict__ attd,
                       MlpPtrs mp, float* __restrict__ scrA) {
  int z = blockIdx.z, b = z >> 2, i = z & 3;
  int mb[4], nav = 0, comb = 0;
  for (int jj = 0; jj < 4; ++jj) { mb[jj] = mask[b * 4 + jj]; nav += mb[jj]; comb = comb * 2 + mb[jj]; }
  if (mb[i] || nav == 0) return;
  int din = nav * DIMC;
  int nTile = blockIdx.x * 64;
  if (nTile >= din) return;
  const float* seg[4] = {nullptr, nullptr, nullptr, nullptr};
  int s = 0;
  for (int jj = 0; jj < 4; ++jj)
    if (mb[jj]) {
      int t = triple_slot(mask, b, i, jj);
      seg[s++] = attd + (size_t)(b * 4 + t) * 262144;
    }
  int rank = 0;
  for (int ii = 0; ii < i; ++ii) if (!mb[ii]) ++rank;
  float* D = scrA + (size_t)b * 1048576 + (size_t)rank * NTOK * din;
  gemm128<A_CONCAT, E_BIAS_GELU>(seg[0], seg[1], seg[2], seg[3], DIMC,
                                 mp.w1[comb], din, D, din, mp.b1[comb],
                                 blockIdx.y * 128, nTile, din);
}

__global__ void k_mlp2(const int* __restrict__ mask, const float* __restrict__ scrA,
                       MlpPtrs mp, float* __restrict__ dout) {
  int z = blockIdx.z, b = z >> 2, i = z & 3;
  int mb[4], nav = 0, comb = 0;
  for (int jj = 0; jj < 4; ++jj) { mb[jj] = mask[b * 4 + jj]; nav += mb[jj]; comb = comb * 2 + mb[jj]; }
  if (mb[i] || nav == 0) return;
  int din = nav * DIMC;
  int rank = 0;
  for (int ii = 0; ii < i; ++ii) if (!mb[ii]) ++rank;
  const float* A = scrA + (size_t)b * 1048576 + (size_t)rank * NTOK * din;
  float* D = dout + (size_t)i * (NB * DIMC * NTOK) + (size_t)b * DIMC * NTOK;
  gemm128<A_ROW, E_BIAS_TSTORE>(A, A, A, A, din, mp.w2[comb], DIMC, D, NTOK, mp.b2[comb],
                                blockIdx.y * 128, blockIdx.x * 64, din);
}

// ---------------------------------------------------------------------------
// Host launch
// ---------------------------------------------------------------------------
extern "C" void kernel_launch(void* const* d_in, const int* in_sizes, int n_in,
                              void* d_out, int out_size, void* d_ws, size_t ws_size,
                              hipStream_t stream) {
  (void)in_sizes; (void)n_in;
  const float* rt[4] = {(const float*)d_in[0], (const float*)d_in[1],
                        (const float*)d_in[2], (const float*)d_in[3]};
  const int*   mask  = (const int*)d_in[4];
  const float* at    = (const float*)d_in[5];
  const float* ucaq  = (const float*)d_in[6];
  const float* ucak  = (const float*)d_in[7];
  const float* ucav  = (const float*)d_in[8];
  const float* ucapw = (const float*)d_in[9];
  const float* ucapb = (const float*)d_in[10];
  const float* acaq  = (const float*)d_in[11];
  const float* acak  = (const float*)d_in[12];
  const float* acav  = (const float*)d_in[13];
  const float* acapw = (const float*)d_in[14];
  const float* aaq   = (const float*)d_in[15];
  const float* aak   = (const float*)d_in[16];
  const float* aav   = (const float*)d_in[17];
  const float* aapw  = (const float*)d_in[18];
  const float* acapb = (const float*)d_in[19];
  const float* aapb  = (const float*)d_in[20];

  MlpPtrs mp = {};
  for (int c = 1; c < 15; ++c) {
    mp.w1[c] = (const float*)d_in[21 + (c - 1) * 4 + 0];
    mp.b1[c] = (const float*)d_in[21 + (c - 1) * 4 + 1];
    mp.w2[c] = (const float*)d_in[21 + (c - 1) * 4 + 2];
    mp.b2[c] = (const float*)d_in[21 + (c - 1) * 4 + 3];
  }

  float* out = (float*)d_out;
  float* ws  = (float*)d_ws;

  // workspace layout (floats)
  size_t off = 0;
  float* ats  = ws + off; off += (size_t)NB * 4 * 8 * DIMC;       // 262144
  float* ac   = ws + off; off += (size_t)NB * 16 * 8 * DIMC;      // 1048576
  float* wsc  = ws + off; off += (size_t)4 * DIMC * 64;           // 131072
  float* T1   = ws + off; off += (size_t)64 * NTOK * 64;          // 2097152
  float* U1   = ws + off; off += (size_t)64 * NTOK * 64;
  float* W2k  = ws + off; off += (size_t)64 * DIMC * 64;
  float* W2q  = ws + off; off += (size_t)64 * DIMC * 64;
  float* Sk   = ws + off; off += (size_t)64 * NTOK * 64;
  float* Sq   = ws + off; off += (size_t)64 * NTOK * 64;
  float* U2   = ws + off; off += (size_t)64 * NTOK * 64;
  float* av2  = ws + off; off += (size_t)64 * 8 * DIMC;           // 262144
  float* scrA = ws + off; off += (size_t)NB * 1048576;            // P / y1 (time-shared)
  float* attd = ws + off; off += (size_t)64 * 262144;             // attended slots
  const size_t needed = off * sizeof(float);                      // ~200 MB

  // pass-through: outputs default to inputs (missing slices overwritten below)
  const size_t tb = (size_t)NB * DIMC * NTOK * sizeof(float);
  for (int i = 0; i < 4; ++i)
    hipMemcpyAsync((char*)d_out + (size_t)i * tb, d_in[i], tb, hipMemcpyDeviceToDevice, stream);

  if (ws_size < needed) {  // not enough scratch: still deterministic (loss=0)
    k_set_loss<<<1, 1, 0, stream>>>(out + (size_t)out_size - 1);
    return;
  }

  // Phase 1: agent-token updates for available modalities (uca)
  k_uca_prep<<<4, 256, 0, stream>>>(at, ucaq, ucak, wsc, out + (size_t)out_size - 1);
  k_uca_scores<<<dim3(1, 4, 64), 256, 0, stream>>>(mask, rt[0], rt[1], rt[2], rt[3], wsc, T1);
  k_uca_softmax<<<64, 64, 0, stream>>>(mask, T1);
  k_uca_U<<<dim3(1, 4, 64), 256, 0, stream>>>(mask, rt[0], rt[1], rt[2], rt[3], T1, U1);
  k_uca_finish<<<64, 256, 0, stream>>>(mask, at, ucav, ucapw, ucapb, U1, ats);

  // Phase 2: agent-agent cross attention (aca)
  k_aca_cross<<<256, 256, 0, stream>>>(mask, ats, acaq, acak, acav, acapw, acapb, ac);

  // Phase 3: agent attention over feature maps (aa)
  k_agent_prep<<<256, 256, 0, stream>>>(mask, ac, aaq, aak, W2q, W2k);
  k_agent_scores<<<dim3(1, 4, 512), 256, 0, stream>>>(mask, rt[0], rt[1], rt[2], rt[3], W2k, W2q, Sk, Sq);
  k_agent_softmax_k<<<256, 64, 0, stream>>>(mask, Sk);
  k_agent_softmax_q<<<256, 256, 0, stream>>>(mask, Sq);
  k_agent_U<<<dim3(1, 4, 256), 256, 0, stream>>>(mask, rt[0], rt[1], rt[2], rt[3], Sk, U2);
  k_agent_av<<<256, 256, 0, stream>>>(mask, aav, U2, av2);
  k_attended_pre<<<256, 256, 0, stream>>>(mask, Sq, av2, scrA);
  k_attended_gemm<<<dim3(8, 4, 256), 256, 0, stream>>>(mask, scrA, aapw, aapb, attd);

  // Phase 4: per-comb MLP reconstruction, transposed store into output layout
  k_mlp1<<<dim3(24, 4, 64), 256, 0, stream>>>(mask, attd, mp, scrA);
  k_mlp2<<<dim3(8, 4, 64), 256, 0, stream>>>(mask, scrA, mp, out);
}